// DAC_structure_37452114821757
// MI455X (gfx1250) — compile-verified
//
#include <hip/hip_runtime.h>
#include <math.h>

typedef __attribute__((ext_vector_type(2))) float v2f;
typedef __attribute__((ext_vector_type(8))) float v8f;

// D = A(16x4 f32) * B(4x16 f32) + C(16x16 f32)   -> v_wmma_f32_16x16x4_f32
__device__ __forceinline__ v8f wmma_f32_16x16x4(v2f a, v2f b, v8f c) {
    return __builtin_amdgcn_wmma_f32_16x16x4_f32(
        /*neg_a=*/false, a, /*neg_b=*/false, b,
        /*c_mod=*/(short)0, c, /*reuse_a=*/false, /*reuse_b=*/false);
}

__device__ __forceinline__ float rmax16(float v) {
    v = fmaxf(v, __shfl_xor(v, 1, 32));
    v = fmaxf(v, __shfl_xor(v, 2, 32));
    v = fmaxf(v, __shfl_xor(v, 4, 32));
    v = fmaxf(v, __shfl_xor(v, 8, 32));
    return v;
}
__device__ __forceinline__ float rsum16(float v) {
    v += __shfl_xor(v, 1, 32);
    v += __shfl_xor(v, 2, 32);
    v += __shfl_xor(v, 4, 32);
    v += __shfl_xor(v, 8, 32);
    return v;
}

// ---------------- Branch 1: patch_size  ------------------------------------
// Q,K: [256, 32, 8, 64].  Per (b,h): S = softmax(Q K^T / 8) (32x32),
// mean over 32 channels, 8x8 element repeat -> out[8,8,256,256].
// grid 64 (= 8 outer * 8 heads), block 256 (8 wave32, 4 channels per wave).
__global__ __launch_bounds__(256) void dac_patch_size_kernel(
    const float* __restrict__ Q, const float* __restrict__ K,
    float* __restrict__ out) {
    __shared__ float sP[8 * 1024];  // per-wave 32x32 partial channel sums

    const int o    = blockIdx.x >> 3;
    const int h    = blockIdx.x & 7;
    const int tid  = threadIdx.x;
    const int w    = tid >> 5;
    const int lane = tid & 31;
    const int hi   = lane >> 4;   // lane half: selects K pair / +8 row offset
    const int ln   = lane & 15;   // A row / B col within tile

    v8f msum[2][2] = {};          // channel-summed softmax tiles (2x2 of 16x16)

    for (int ci = 0; ci < 4; ++ci) {
        const int b = o * 32 + w * 4 + ci;           // channel innermost in b
        v8f acc[2][2] = {};
        const float* qb = Q + (((size_t)b * 32) * 8 + h) * 64;
        const float* kb = K + (((size_t)b * 32) * 8 + h) * 64;
#pragma unroll
        for (int kk = 0; kk < 16; ++kk) {
            const int ke = kk * 4 + 2 * hi;          // e index of this K-pair
            v2f a0 = *(const v2f*)(qb + (size_t)ln * 512 + ke);          // rows 0..15
            v2f a1 = *(const v2f*)(qb + (size_t)(ln + 16) * 512 + ke);   // rows 16..31
            v2f b0 = *(const v2f*)(kb + (size_t)ln * 512 + ke);          // cols 0..15
            v2f b1 = *(const v2f*)(kb + (size_t)(ln + 16) * 512 + ke);   // cols 16..31
            acc[0][0] = wmma_f32_16x16x4(a0, b0, acc[0][0]);
            acc[0][1] = wmma_f32_16x16x4(a0, b1, acc[0][1]);
            acc[1][0] = wmma_f32_16x16x4(a1, b0, acc[1][0]);
            acc[1][1] = wmma_f32_16x16x4(a1, b1, acc[1][1]);
        }
        // scale (1/sqrt(64)) + row softmax. Row M = r + 8*hi + 16*mt lives in
        // one 16-lane half; its 32 cols are (lane&15)+16*nt of that half.
#pragma unroll
        for (int mt = 0; mt < 2; ++mt) {
#pragma unroll
            for (int r = 0; r < 8; ++r) {
                float x0 = acc[mt][0][r] * 0.125f;
                float x1 = acc[mt][1][r] * 0.125f;
                float mx = rmax16(fmaxf(x0, x1));
                float e0 = __expf(x0 - mx);
                float e1 = __expf(x1 - mx);
                float s  = rsum16(e0 + e1);
                float is = 1.0f / s;
                msum[mt][0][r] += e0 * is;
                msum[mt][1][r] += e1 * is;
            }
        }
    }

    // store per-wave partial 32x32 (sum of its 4 channels)
    float* myP = sP + w * 1024;
#pragma unroll
    for (int mt = 0; mt < 2; ++mt)
#pragma unroll
        for (int nt = 0; nt < 2; ++nt)
#pragma unroll
            for (int r = 0; r < 8; ++r)
                myP[(r + 8 * hi + 16 * mt) * 32 + ln + 16 * nt] = msum[mt][nt][r];
    __syncthreads();

    // deterministic fixed-order reduction over 8 waves (each index owned by 1 thread)
    for (int i = tid; i < 1024; i += 256) {
        float s = 0.f;
#pragma unroll
        for (int w2 = 0; w2 < 8; ++w2) s += sP[w2 * 1024 + i];
        sP[i] = s;
    }
    __syncthreads();

    // broadcast-write: out[o,h,L,S] = mean[L/8, S/8]; coalesced stores
    const size_t base = (size_t)(o * 8 + h) * 65536u;
    for (int i = tid; i < 65536; i += 256) {
        int L = i >> 8, S = i & 255;
        out[base + i] = sP[(L >> 3) * 32 + (S >> 3)] * (1.0f / 32.0f);
    }
}

// ---------------- Branch 2: patch_num --------------------------------------
// Q,K: [256, 8, 8, 64]. Per (b,h): 8x8 scores in one zero-padded 16x16 WMMA
// tile, masked softmax, channel mean, 32x32 tile -> out[8,8,256,256].
__global__ __launch_bounds__(256) void dac_patch_num_kernel(
    const float* __restrict__ Q, const float* __restrict__ K,
    float* __restrict__ out) {
    __shared__ float sP[8 * 64];

    const int o    = blockIdx.x >> 3;
    const int h    = blockIdx.x & 7;
    const int tid  = threadIdx.x;
    const int w    = tid >> 5;
    const int lane = tid & 31;
    const int hi   = lane >> 4;
    const int ln   = lane & 15;
    const bool vl  = (ln < 8);    // valid A-row / B-col (pad rows/cols 8..15)

    float msum[8] = {};

    for (int ci = 0; ci < 4; ++ci) {
        const int b = o * 32 + w * 4 + ci;
        v8f acc = {};
        const float* qb = Q + (((size_t)b * 8) * 8 + h) * 64;
        const float* kb = K + (((size_t)b * 8) * 8 + h) * 64;
        const v2f z = {0.f, 0.f};
#pragma unroll
        for (int kk = 0; kk < 16; ++kk) {
            const int ke = kk * 4 + 2 * hi;
            v2f a = vl ? *(const v2f*)(qb + (size_t)ln * 512 + ke) : z;
            v2f bb = vl ? *(const v2f*)(kb + (size_t)ln * 512 + ke) : z;
            acc = wmma_f32_16x16x4(a, bb, acc);
        }
#pragma unroll
        for (int r = 0; r < 8; ++r) {
            float x  = acc[r] * 0.125f;
            float mx = rmax16(vl ? x : -INFINITY);
            float e  = vl ? __expf(x - mx) : 0.f;
            float s  = rsum16(e);
            msum[r] += e / s;   // garbage for rows M>=8 / cols>=8, never stored
        }
    }

    // only lanes 0..7 (hi==0, col<8) hold valid rows 0..7
    if (hi == 0 && vl) {
#pragma unroll
        for (int r = 0; r < 8; ++r) sP[w * 64 + r * 8 + ln] = msum[r];
    }
    __syncthreads();

    if (tid < 64) {
        float s = 0.f;
#pragma unroll
        for (int w2 = 0; w2 < 8; ++w2) s += sP[w2 * 64 + tid];
        sP[tid] = s;
    }
    __syncthreads();

    const size_t base = (size_t)(o * 8 + h) * 65536u;
    for (int i = tid; i < 65536; i += 256) {
        int L = i >> 8, S = i & 255;
        out[base + i] = sP[(L & 7) * 8 + (S & 7)] * (1.0f / 32.0f);
    }
}

extern "C" void kernel_launch(void* const* d_in, const int* in_sizes, int n_in,
                              void* d_out, int out_size, void* d_ws, size_t ws_size,
                              hipStream_t stream) {
    const float* q_ps = (const float*)d_in[0];  // [256,32,8,64]
    const float* k_ps = (const float*)d_in[1];  // [256,32,8,64]
    const float* q_pn = (const float*)d_in[2];  // [256,8,8,64]
    const float* k_pn = (const float*)d_in[3];  // [256,8,8,64]
    float* out = (float*)d_out;                 // [2, 8,8,256,256] flat

    float* out_ps = out;                        // first tuple element
    float* out_pn = out + (size_t)8 * 8 * 256 * 256;  // second tuple element

    dac_patch_size_kernel<<<64, 256, 0, stream>>>(q_ps, k_ps, out_ps);
    dac_patch_num_kernel<<<64, 256, 0, stream>>>(q_pn, k_pn, out_pn);
}